// GraphAttentionEdgeConv_63170378989710
// MI455X (gfx1250) — compile-verified
//
#include <hip/hip_runtime.h>

typedef float v2f __attribute__((ext_vector_type(2)));
typedef float v4f __attribute__((ext_vector_type(4)));
typedef float v8f __attribute__((ext_vector_type(8)));

#define DD 256
#define NNODES 4096
#define NEDGES 32768
#define NSLICE 16
#define ESLICE (NEDGES / NSLICE)   /* 2048 edges per slice        */
#define ECHUNK 32                  /* edges staged per iteration  */
#define NCHUNK (ESLICE / ECHUNK)   /* 64 double-buffered chunks   */
#define LDS_PITCH 260              /* 256 + 4: bank-conflict-free, 16B aligned */

// ---------------------------------------------------------------------------
// CDNA5 async global->LDS copy (ASYNCcnt path), 16 bytes per lane.
// LDS destination address = low 32 bits of the flat pointer (aperture rule).
// ---------------------------------------------------------------------------
__device__ __forceinline__ void async_copy_f4(const float* g, float* l) {
  const unsigned lds_addr = (unsigned)(unsigned long long)(void*)l;
  asm volatile("global_load_async_to_lds_b128 %0, %1, off"
               :: "v"(lds_addr), "v"(g)
               : "memory");
}

__device__ __forceinline__ void wait_async0() {
  asm volatile("s_wait_asynccnt 0x0" ::: "memory");
}

// ---------------------------------------------------------------------------
// Kernel 1: Y' = Y @ W^T   (E x D x D GEMM, f32 WMMA 16x16x4)
// One wave per 16x16 output tile.
// ---------------------------------------------------------------------------
__global__ __launch_bounds__(32) void k_yprime(const float* __restrict__ Y,
                                               const float* __restrict__ W,
                                               float* __restrict__ Yp) {
  const int lane = threadIdx.x & 31;
  const int m0 = blockIdx.x * 16;      // edge-row tile
  const int n0 = blockIdx.y * 16;      // output-col tile
  const int r   = lane & 15;
  const int khi = (lane >> 4) * 2;     // k sub-offset per half-wave

  const float* arow = Y + (size_t)(m0 + r) * DD + khi;   // A = Y tile (16 x 4)
  const float* brow = W + (size_t)(n0 + r) * DD + khi;   // B[k,n] = W[n,k]

  v8f c = {};
#pragma unroll
  for (int kk = 0; kk < DD / 4; ++kk) {
    v2f a = *(const v2f*)(arow + kk * 4);
    v2f b = *(const v2f*)(brow + kk * 4);
    c = __builtin_amdgcn_wmma_f32_16x16x4_f32(false, a, false, b, (short)0, c,
                                              false, false);
  }

  const int rbase = (lane >> 4) * 8;   // C: vgpr v -> row v (lanes 0-15) / v+8 (16-31)
#pragma unroll
  for (int v = 0; v < 8; ++v)
    Yp[(size_t)(m0 + rbase + v) * DD + n0 + (lane & 15)] = c[v];
}

// ---------------------------------------------------------------------------
// Kernel 2: X' = X  (initialize residual)
// ---------------------------------------------------------------------------
__global__ __launch_bounds__(256) void k_copyX(const float* __restrict__ X,
                                               float* __restrict__ Xp) {
  const size_t i = (size_t)(blockIdx.x * blockDim.x + threadIdx.x) * 4;
  *(v4f*)(Xp + i) = *(const v4f*)(X + i);
}

// ---------------------------------------------------------------------------
// Kernel 3: flash-style row (max, sumexp) of scores = X @ Y'^T over an E-slice.
// Block = 4 waves; each wave owns a 16-node tile. Y' chunks (32 edges) are
// double-buffered in LDS via async global->LDS loads overlapped with WMMAs.
// ---------------------------------------------------------------------------
__global__ __launch_bounds__(128) void k_scores(const float* __restrict__ X,
                                                const float* __restrict__ Yp,
                                                float* __restrict__ mpart,
                                                float* __restrict__ lpart) {
  __shared__ float lds[2][ECHUNK * LDS_PITCH];
  const int tid   = threadIdx.x;
  const int lane  = tid & 31;
  const int wave  = tid >> 5;
  const int node0 = (blockIdx.x * 4 + wave) * 16;
  const int slice = blockIdx.y;
  const int e0    = slice * ESLICE;
  const int r     = lane & 15;
  const int khi   = (lane >> 4) * 2;

  // Preload X tile A-fragments (constant over all chunks)
  v2f a_frag[DD / 4];
  const float* xrow = X + (size_t)(node0 + r) * DD + khi;
#pragma unroll
  for (int kk = 0; kk < DD / 4; ++kk) a_frag[kk] = *(const v2f*)(xrow + kk * 4);

  float mrun[8], lrun[8];
#pragma unroll
  for (int v = 0; v < 8; ++v) { mrun[v] = -__builtin_inff(); lrun[v] = 0.f; }

  // Staging map: 32 rows x 256 f32 per chunk, 128 threads -> 64 f32 each.
  const int srow = tid >> 2;          // 0..31: staged edge row
  const int sseg = (tid & 3) * 64;    // 64-float segment of that row

  auto stage = [&](int ch, int buf) {
    const int e = e0 + ch * ECHUNK;
    const float* gsrc = Yp + (size_t)(e + srow) * DD + sseg;
    float* ldst = &lds[buf][srow * LDS_PITCH + sseg];
#pragma unroll
    for (int j = 0; j < 16; ++j)
      async_copy_f4(gsrc + j * 4, ldst + j * 4);
  };

  stage(0, 0);
  for (int ch = 0; ch < NCHUNK; ++ch) {
    wait_async0();        // my async writes into the current buffer are done
    __syncthreads();      // everyone's are done; prev buffer no longer read
    if (ch + 1 < NCHUNK) stage(ch + 1, (ch + 1) & 1);  // overlap with compute

    // Two 16x16 score tiles vs the same X tile, K=256 in 64 WMMA steps each
    const float* base  = &lds[ch & 1][0];
    const float* brow0 = base + r * LDS_PITCH + khi;
    const float* brow1 = base + (16 + r) * LDS_PITCH + khi;
    v8f c0 = {}, c1 = {};
#pragma unroll
    for (int kk = 0; kk < DD / 4; ++kk) {
      v2f b0 = *(const v2f*)(brow0 + kk * 4);
      v2f b1 = *(const v2f*)(brow1 + kk * 4);
      c0 = __builtin_amdgcn_wmma_f32_16x16x4_f32(false, a_frag[kk], false, b0,
                                                 (short)0, c0, false, false);
      c1 = __builtin_amdgcn_wmma_f32_16x16x4_f32(false, a_frag[kk], false, b1,
                                                 (short)0, c1, false, false);
    }
    // Online softmax update (per lane: 8 rows, this lane's columns)
#pragma unroll
    for (int v = 0; v < 8; ++v) {
      float s  = c0[v];
      float mo = mrun[v];
      float mn = fmaxf(mo, s);
      lrun[v] = lrun[v] * __expf(mo - mn) + __expf(s - mn);
      mrun[v] = mn;
      s  = c1[v];
      mo = mrun[v];
      mn = fmaxf(mo, s);
      lrun[v] = lrun[v] * __expf(mo - mn) + __expf(s - mn);
      mrun[v] = mn;
    }
  }

  // Reduce the 16 column-lanes of each half-wave (rows split across halves)
#pragma unroll
  for (int off = 1; off < 16; off <<= 1) {
#pragma unroll
    for (int v = 0; v < 8; ++v) {
      const float mo = __shfl_xor(mrun[v], off, 32);
      const float lo = __shfl_xor(lrun[v], off, 32);
      const float mn = fmaxf(mrun[v], mo);
      lrun[v] = lrun[v] * __expf(mrun[v] - mn) + lo * __expf(mo - mn);
      mrun[v] = mn;
    }
  }
  if (r == 0) {
    const int rbase = (lane >> 4) * 8;
#pragma unroll
    for (int v = 0; v < 8; ++v) {
      const int node = node0 + rbase + v;
      mpart[node * NSLICE + slice] = mrun[v];
      lpart[node * NSLICE + slice] = lrun[v];
    }
  }
}

// ---------------------------------------------------------------------------
// Kernel 4: combine slice partials -> per-node max m[n] and denom Z[n]
// ---------------------------------------------------------------------------
__global__ __launch_bounds__(256) void k_combine(const float* __restrict__ mpart,
                                                 const float* __restrict__ lpart,
                                                 float* __restrict__ mfin,
                                                 float* __restrict__ zfin) {
  const int n = blockIdx.x * blockDim.x + threadIdx.x;
  if (n >= NNODES) return;
  float m = -__builtin_inff();
#pragma unroll
  for (int s = 0; s < NSLICE; ++s) m = fmaxf(m, mpart[n * NSLICE + s]);
  float z = 0.f;
#pragma unroll
  for (int s = 0; s < NSLICE; ++s)
    z += lpart[n * NSLICE + s] * __expf(mpart[n * NSLICE + s] - m);
  mfin[n] = m;
  zfin[n] = z;
}

// ---------------------------------------------------------------------------
// Kernel 5: sparse scatter. One wave per edge: recompute endpoint scores,
// weight = exp(s - m)/Z, atomically accumulate w * y'_e into X'.
// ---------------------------------------------------------------------------
__global__ __launch_bounds__(256) void k_edges(const float* __restrict__ X,
                                               const float* __restrict__ Yp,
                                               const long long* __restrict__ eidx,
                                               const float* __restrict__ mfin,
                                               const float* __restrict__ zfin,
                                               float* __restrict__ Xp) {
  const int lane = threadIdx.x & 31;
  const int wave = threadIdx.x >> 5;
  const int e = blockIdx.x * 8 + wave;
  const int a = (int)eidx[(size_t)e * 2 + 0];
  const int b = (int)eidx[(size_t)e * 2 + 1];

  float yv[8];
  float pa = 0.f, pb = 0.f;
#pragma unroll
  for (int j = 0; j < 8; ++j) {
    const int col = lane + j * 32;
    const float y = Yp[(size_t)e * DD + col];
    yv[j] = y;
    pa += X[(size_t)a * DD + col] * y;
    pb += X[(size_t)b * DD + col] * y;
  }
#pragma unroll
  for (int off = 1; off < 32; off <<= 1) {
    pa += __shfl_xor(pa, off, 32);
    pb += __shfl_xor(pb, off, 32);
  }
  const float wa = __expf(pa - mfin[a]) / zfin[a];
  const float wb = __expf(pb - mfin[b]) / zfin[b];
#pragma unroll
  for (int j = 0; j < 8; ++j) {
    const int col = lane + j * 32;
    atomicAdd(&Xp[(size_t)a * DD + col], wa * yv[j]);
    if (b != a) atomicAdd(&Xp[(size_t)b * DD + col], wb * yv[j]);
  }
}

// ---------------------------------------------------------------------------
extern "C" void kernel_launch(void* const* d_in, const int* in_sizes, int n_in,
                              void* d_out, int out_size, void* d_ws, size_t ws_size,
                              hipStream_t stream) {
  const float*     X    = (const float*)d_in[0];
  const float*     Y    = (const float*)d_in[1];
  const long long* eidx = (const long long*)d_in[2];
  const float*     W    = (const float*)d_in[3];

  float* Xp = (float*)d_out;                         // [N, D]
  float* Yp = (float*)d_out + (size_t)NNODES * DD;   // [E, D]

  float* mpart = (float*)d_ws;                       // [N, NSLICE]
  float* lpart = mpart + NNODES * NSLICE;            // [N, NSLICE]
  float* mfin  = lpart + NNODES * NSLICE;            // [N]
  float* zfin  = mfin + NNODES;                      // [N]

  // 1) Y' = Y @ W^T (writes into d_out's Y' region)
  k_yprime<<<dim3(NEDGES / 16, DD / 16), 32, 0, stream>>>(Y, W, Yp);
  // 2) X' = X
  k_copyX<<<(NNODES * DD / 4) / 256, 256, 0, stream>>>(X, Xp);
  // 3) per-(node, slice) softmax partials over scores = X @ Y'^T
  k_scores<<<dim3(NNODES / 64, NSLICE), 128, 0, stream>>>(X, Yp, mpart, lpart);
  // 4) combine to per-node (m, Z)
  k_combine<<<NNODES / 256, 256, 0, stream>>>(mpart, lpart, mfin, zfin);
  // 5) masked attention scatter: X' += exp(s - m)/Z * Y'
  k_edges<<<NEDGES / 8, 256, 0, stream>>>(X, Yp, eidx, mfin, zfin, Xp);
}